// PointPillarFeatureNet_60533269069953
// MI455X (gfx1250) — compile-verified
//
#include <hip/hip_runtime.h>

typedef float v2f __attribute__((ext_vector_type(2)));
typedef float v8f __attribute__((ext_vector_type(8)));

#define VXY   0.16f
#define XOFF  0.08f     // VX/2 + 0.0
#define YOFF  -39.6f    // VY/2 - 39.68
#define NPTS  32
#define COUT  64

// ---------------------------------------------------------------------------
// Kernel 0: zero the 54-float stats accumulator region (harness poisons ws).
// ---------------------------------------------------------------------------
__global__ void pfn_k0_zero(float* __restrict__ acc) {
  acc[threadIdx.x] = 0.0f;
}

// ---------------------------------------------------------------------------
// Kernel 1: accumulate m = sum(feats) (9) and S = sum(feats^T feats) (45, upper
// triangular). One wave per pillar per iteration; lane == point index.
// acc[0..44] = S (row-major upper triangle), acc[45..53] = m.
// ---------------------------------------------------------------------------
__global__ __launch_bounds__(256) void pfn_k1_stats(
    const float* __restrict__ inputs, const int* __restrict__ num_points,
    const int* __restrict__ coords, float* __restrict__ acc, int P) {
  const int lane   = threadIdx.x & 31;
  const int wave   = threadIdx.x >> 5;
  const int gwave  = blockIdx.x * (blockDim.x >> 5) + wave;
  const int nwaves = gridDim.x * (blockDim.x >> 5);

  float s[45];
  float m[9];
#pragma unroll
  for (int i = 0; i < 45; ++i) s[i] = 0.0f;
#pragma unroll
  for (int i = 0; i < 9; ++i) m[i] = 0.0f;

  for (int p = gwave; p < P; p += nwaves) {
    const float4 pt = reinterpret_cast<const float4*>(inputs)[p * NPTS + lane];
    const int   np  = num_points[p];
    const float cx  = (float)coords[p * 4 + 3] * VXY + XOFF;
    const float cy  = (float)coords[p * 4 + 2] * VXY + YOFF;

    // cluster mean: sum xyz over all 32 slots, divide by num_points
    float sx = pt.x, sy = pt.y, sz = pt.z;
#pragma unroll
    for (int off = 16; off > 0; off >>= 1) {
      sx += __shfl_xor(sx, off);
      sy += __shfl_xor(sy, off);
      sz += __shfl_xor(sz, off);
    }
    const float inv = 1.0f / (float)np;
    const float msk = (lane < np) ? 1.0f : 0.0f;

    float f[9];
    f[0] = pt.x * msk;
    f[1] = pt.y * msk;
    f[2] = pt.z * msk;
    f[3] = pt.w * msk;
    f[4] = (pt.x - sx * inv) * msk;
    f[5] = (pt.y - sy * inv) * msk;
    f[6] = (pt.z - sz * inv) * msk;
    f[7] = (pt.x - cx) * msk;
    f[8] = (pt.y - cy) * msk;

    int idx = 0;
#pragma unroll
    for (int i = 0; i < 9; ++i) {
      m[i] += f[i];
#pragma unroll
      for (int j = i; j < 9; ++j) s[idx++] += f[i] * f[j];
    }
  }

  // wave reduction of the 54 partials
#pragma unroll
  for (int i = 0; i < 45; ++i) {
#pragma unroll
    for (int off = 16; off > 0; off >>= 1) s[i] += __shfl_xor(s[i], off);
  }
#pragma unroll
  for (int i = 0; i < 9; ++i) {
#pragma unroll
    for (int off = 16; off > 0; off >>= 1) m[i] += __shfl_xor(m[i], off);
  }

  // block reduction via LDS float atomics, then one global atomic per value
  __shared__ float red[54];
  if (threadIdx.x < 54) red[threadIdx.x] = 0.0f;
  __syncthreads();
  if (lane == 0) {
#pragma unroll
    for (int i = 0; i < 45; ++i) atomicAdd(&red[i], s[i]);
#pragma unroll
    for (int i = 0; i < 9; ++i) atomicAdd(&red[45 + i], m[i]);
  }
  __syncthreads();
  if (threadIdx.x < 54) unsafeAtomicAdd(&acc[threadIdx.x], red[threadIdx.x]);
}

// ---------------------------------------------------------------------------
// Kernel 2: fold BatchNorm into weights/bias.
// mean_d = (m . W[:,d]) / PN ;  E[x^2]_d = W[:,d]^T S W[:,d] / PN
// Wp (12x64, rows 9..11 zero) = W * gamma * rsqrt(var+eps) ; bias = beta - mean*scale
// ---------------------------------------------------------------------------
__global__ void pfn_k2_fold(const float* __restrict__ acc,
                            const float* __restrict__ W,
                            const float* __restrict__ gamma,
                            const float* __restrict__ beta,
                            float* __restrict__ Wp, float* __restrict__ bias,
                            float invPN) {
  const int d = threadIdx.x;  // 64 threads
  float w[9];
#pragma unroll
  for (int c = 0; c < 9; ++c) w[c] = W[c * COUT + d];

  float mean = 0.0f;
#pragma unroll
  for (int c = 0; c < 9; ++c) mean += acc[45 + c] * w[c];
  mean *= invPN;

  float e2 = 0.0f;
  int idx = 0;
#pragma unroll
  for (int i = 0; i < 9; ++i) {
#pragma unroll
    for (int j = i; j < 9; ++j) {
      const float t = acc[idx++] * w[i] * w[j];
      e2 += (i == j) ? t : 2.0f * t;
    }
  }
  e2 *= invPN;

  const float var   = e2 - mean * mean;
  const float scale = gamma[d] * rsqrtf(var + 1.0e-3f);
#pragma unroll
  for (int c = 0; c < 9; ++c) Wp[c * COUT + d] = w[c] * scale;
#pragma unroll
  for (int c = 9; c < 12; ++c) Wp[c * COUT + d] = 0.0f;
  bias[d] = beta[d] - mean * scale;
}

// ---------------------------------------------------------------------------
// Kernel 3: one wave per pillar. Build 9 features -> LDS (stride 12, K padded
// to 12), then 2x4x3 V_WMMA_F32_16X16X4_F32 for the 32x64 tile, fused
// +bias / ReLU / max-over-points epilogue.
//
// fp32 WMMA layouts (ISA 7.12.2):
//  A 16x4 : VGPR0 = {lanes0-15: K=0, lanes16-31: K=2}, VGPR1 = {K=1 | K=3}
//  B 4x16 : VGPR0 = {lanes0-15: K=0 N=lane, lanes16-31: K=2}, VGPR1 = {K=1|K=3}
//  C 16x16: VGPR r = {lanes0-15: M=r, lanes16-31: M=r+8}, N = lane%16
//  (C's M permutation is irrelevant here: we reduce with max over M.)
// ---------------------------------------------------------------------------
__global__ __launch_bounds__(128) void pfn_k3_gemm(
    const float* __restrict__ inputs, const int* __restrict__ num_points,
    const int* __restrict__ coords, const float* __restrict__ Wp,
    const float* __restrict__ bias, float* __restrict__ out, int P) {
  __shared__ float lds[4][NPTS * 12];
  const int lane   = threadIdx.x & 31;
  const int wave   = threadIdx.x >> 5;
  const int gwave  = blockIdx.x * 4 + wave;
  const int nwaves = gridDim.x * 4;
  float* feat = lds[wave];

  const int csub = lane & 15;        // column within 16-wide tile / row within M-tile
  const int rlo  = (lane >> 4) * 2;  // K offset inside a K=4 group (0 or 2)

  // B fragments: folded weights, resident for the whole grid-stride loop.
  v2f bfrag[4][3];
#pragma unroll
  for (int nt = 0; nt < 4; ++nt) {
#pragma unroll
    for (int ks = 0; ks < 3; ++ks) {
      const int row = ks * 4 + rlo;
      bfrag[nt][ks].x = Wp[row * COUT + nt * 16 + csub];
      bfrag[nt][ks].y = Wp[(row + 1) * COUT + nt * 16 + csub];
    }
  }
  float biasv[4];
#pragma unroll
  for (int nt = 0; nt < 4; ++nt) biasv[nt] = bias[nt * 16 + csub];

  for (int p = gwave; p < P; p += nwaves) {
    if (p + nwaves < P)
      __builtin_prefetch(&inputs[(size_t)(p + nwaves) * NPTS * 4], 0, 0);

    const float4 pt = reinterpret_cast<const float4*>(inputs)[p * NPTS + lane];
    const int   np  = num_points[p];
    const float cx  = (float)coords[p * 4 + 3] * VXY + XOFF;
    const float cy  = (float)coords[p * 4 + 2] * VXY + YOFF;

    float sx = pt.x, sy = pt.y, sz = pt.z;
#pragma unroll
    for (int off = 16; off > 0; off >>= 1) {
      sx += __shfl_xor(sx, off);
      sy += __shfl_xor(sy, off);
      sz += __shfl_xor(sz, off);
    }
    const float inv = 1.0f / (float)np;
    const float msk = (lane < np) ? 1.0f : 0.0f;

    // stage padded 12-float feature row for point `lane`
    float4* fp = reinterpret_cast<float4*>(&feat[lane * 12]);
    fp[0] = make_float4(pt.x * msk, pt.y * msk, pt.z * msk, pt.w * msk);
    fp[1] = make_float4((pt.x - sx * inv) * msk, (pt.y - sy * inv) * msk,
                        (pt.z - sz * inv) * msk, (pt.x - cx) * msk);
    fp[2] = make_float4((pt.y - cy) * msk, 0.0f, 0.0f, 0.0f);
    // Same-wave DS ops are hardware in-order; just stop compiler reordering.
    asm volatile("" ::: "memory");

    v8f accv[2][4];
#pragma unroll
    for (int mt = 0; mt < 2; ++mt) {
      const int prow = mt * 16 + csub;
      v2f a[3];
#pragma unroll
      for (int ks = 0; ks < 3; ++ks) {
        const float2 t =
            *reinterpret_cast<const float2*>(&feat[prow * 12 + ks * 4 + rlo]);
        a[ks].x = t.x;
        a[ks].y = t.y;
      }
#pragma unroll
      for (int nt = 0; nt < 4; ++nt) {
        v8f c = {};
        c = __builtin_amdgcn_wmma_f32_16x16x4_f32(false, a[0], false,
                                                  bfrag[nt][0], (short)0, c,
                                                  false, false);
        c = __builtin_amdgcn_wmma_f32_16x16x4_f32(false, a[1], false,
                                                  bfrag[nt][1], (short)0, c,
                                                  false, false);
        c = __builtin_amdgcn_wmma_f32_16x16x4_f32(false, a[2], false,
                                                  bfrag[nt][2], (short)0, c,
                                                  false, false);
        accv[mt][nt] = c;
      }
    }
    asm volatile("" ::: "memory");

    // epilogue: +bias, ReLU (via max with 0), max over all 32 points
#pragma unroll
    for (int nt = 0; nt < 4; ++nt) {
      float vmax = 0.0f;  // ReLU floor
#pragma unroll
      for (int r = 0; r < 8; ++r) {
        vmax = fmaxf(vmax, accv[0][nt][r] + biasv[nt]);
        vmax = fmaxf(vmax, accv[1][nt][r] + biasv[nt]);
      }
      vmax = fmaxf(vmax, __shfl_xor(vmax, 16));
      if (lane < 16) out[p * COUT + nt * 16 + lane] = vmax;
    }
  }
}

// ---------------------------------------------------------------------------
extern "C" void kernel_launch(void* const* d_in, const int* in_sizes, int n_in,
                              void* d_out, int out_size, void* d_ws,
                              size_t ws_size, hipStream_t stream) {
  const float* inputs     = (const float*)d_in[0];
  const int*   num_points = (const int*)d_in[1];
  const int*   coords     = (const int*)d_in[2];
  const float* W          = (const float*)d_in[3];
  const float* gamma      = (const float*)d_in[4];
  const float* beta       = (const float*)d_in[5];
  float*       out        = (float*)d_out;
  const int    P          = in_sizes[1];  // num_points length == #pillars

  // workspace layout (floats): [0..63] stats acc, [64..831] Wp(12x64), [832..895] bias
  float* acc  = (float*)d_ws;
  float* Wp   = acc + 64;
  float* bias = Wp + 12 * COUT;

  pfn_k0_zero<<<1, 64, 0, stream>>>(acc);
  pfn_k1_stats<<<256, 256, 0, stream>>>(inputs, num_points, coords, acc, P);
  const float invPN = 1.0f / (float)(P * NPTS);
  pfn_k2_fold<<<1, 64, 0, stream>>>(acc, W, gamma, beta, Wp, bias, invPN);

  int blocks = (P + 3) / 4;
  if (blocks > 480) blocks = 480;
  pfn_k3_gemm<<<blocks, 128, 0, stream>>>(inputs, num_points, coords, Wp, bias,
                                          out, P);
}